// XSReLU_perc_47528108097992
// MI455X (gfx1250) — compile-verified
//
#include <hip/hip_runtime.h>
#include <hip/hip_bf16.h>
#include <stdint.h>

// ---------------------------------------------------------------------------
// XSReLU-percentile for MI455X (gfx1250)
//   per row: cutoff = exact 80th percentile (rank 52428 of 65536, ascending)
//   out     = relu(in - cutoff)
// Row (256KB) is async-DMA'd once into LDS (CDNA5: 320KB/WGP), exact cutoff
// via 4x 8-bit radix-select on monotonic u32 keys. Pass 0 uses per-wave
// privatized, bank-skewed histograms (exponent-clustered data would otherwise
// serialize LDS atomics); pass 0 is software-pipelined against the async DMA
// using in-order ASYNCcnt semantics. HBM traffic = 1x read + 1x write.
// ---------------------------------------------------------------------------

#define N_COLS      65536
#define K_RANK      52428        // int(65536 * 0.8), 0-indexed ascending rank
#define BLOCK       1024
#define CHUNKS      (N_COLS / 4) // 16-byte chunks per row (16384)
#define ITERS       (CHUNKS / BLOCK)          // 16
#define HALF_ITERS  (ITERS / 2)               // 8
#define NWAVES      (BLOCK / 32)              // 32 (wave32)
#define HIST_STRIDE 257          // 257 % 64 == 1 -> wave w starts on bank w

#if defined(__has_builtin)
#  if __has_builtin(__builtin_amdgcn_global_load_async_to_lds_b128)
#    define HAVE_ASYNC_LDS 1
#  endif
#  if __has_builtin(__builtin_amdgcn_s_wait_asynccnt)
#    define HAVE_WAIT_ASYNC 1
#  endif
#endif
#ifndef HAVE_ASYNC_LDS
#  define HAVE_ASYNC_LDS 0
#endif
#ifndef HAVE_WAIT_ASYNC
#  define HAVE_WAIT_ASYNC 0
#endif

typedef int   v4i __attribute__((ext_vector_type(4)));
typedef float v4f __attribute__((ext_vector_type(4)));
typedef __attribute__((address_space(1))) v4i* gv4i_ptr;  // global int4*
typedef __attribute__((address_space(3))) v4i* lv4i_ptr;  // LDS int4*

template <int N>
__device__ __forceinline__ void wait_async_le() {
#if HAVE_ASYNC_LDS
#  if HAVE_WAIT_ASYNC
  __builtin_amdgcn_s_wait_asynccnt(N);
#  else
  asm volatile("s_wait_asynccnt %0" :: "i"(N) : "memory");
#  endif
#endif
}

// monotonic order-preserving map: float bits -> unsigned key (ascending)
__device__ __forceinline__ uint32_t mono_key(uint32_t u) {
  return (u & 0x80000000u) ? ~u : (u | 0x80000000u);
}

extern "C" __global__ __launch_bounds__(BLOCK, 1)
void xsrelu_perc_kernel(const float* __restrict__ in, float* __restrict__ out) {
  extern __shared__ uint32_t smem[];
  uint32_t* raw     = smem;                          // [65536] row bits
  uint32_t* whist   = smem + N_COLS;                 // [32*257] pass-0 per-wave
  uint32_t* hist    = whist + NWAVES * HIST_STRIDE;  // [256] shared histogram
  uint32_t* s_state = hist + 256;                    // [0]=prefix, [1]=k

  const int tid = threadIdx.x;
  const int wid = tid >> 5;                          // wave id (wave32)
  const long long rowBase = (long long)blockIdx.x * (long long)N_COLS;
  const float* src = in + rowBase;
  float* dst = out + rowBase;

  // ---- Stage 1: row -> LDS via async DMA (two halves, in-order ASYNCcnt) --
#pragma unroll
  for (int j = 0; j < ITERS; ++j) {
    const int c = tid + j * BLOCK;                   // 16B chunk index
#if HAVE_ASYNC_LDS
    __builtin_amdgcn_global_load_async_to_lds_b128(
        (gv4i_ptr)(src + 4 * c), (lv4i_ptr)(raw + 4 * c), 0, 0);
#else
    const uint4 v = reinterpret_cast<const uint4*>(src)[c];
    reinterpret_cast<uint4*>(raw)[c] = v;
#endif
  }

  // zero the privatized pass-0 histograms while the DMA streams
  for (int i = tid; i < NWAVES * HIST_STRIDE; i += BLOCK) whist[i] = 0u;

  // ---- Pass 0 (top byte), privatized + pipelined against the DMA ---------
  uint32_t* mybins = whist + wid * HIST_STRIDE;

  wait_async_le<HALF_ITERS>();   // first 8 async ops (half A) complete
  __syncthreads();
#pragma unroll 4
  for (int j = 0; j < HALF_ITERS; ++j) {
    const int c = tid + j * BLOCK;
    const uint4 q = reinterpret_cast<uint4*>(raw)[c];
    atomicAdd(&mybins[mono_key(q.x) >> 24], 1u);
    atomicAdd(&mybins[mono_key(q.y) >> 24], 1u);
    atomicAdd(&mybins[mono_key(q.z) >> 24], 1u);
    atomicAdd(&mybins[mono_key(q.w) >> 24], 1u);
  }
  wait_async_le<0>();            // half B complete
  __syncthreads();
#pragma unroll 4
  for (int j = HALF_ITERS; j < ITERS; ++j) {
    const int c = tid + j * BLOCK;
    const uint4 q = reinterpret_cast<uint4*>(raw)[c];
    atomicAdd(&mybins[mono_key(q.x) >> 24], 1u);
    atomicAdd(&mybins[mono_key(q.y) >> 24], 1u);
    atomicAdd(&mybins[mono_key(q.z) >> 24], 1u);
    atomicAdd(&mybins[mono_key(q.w) >> 24], 1u);
  }
  __syncthreads();

  // reduce 32 privatized histograms -> hist[256]
  if (tid < 256) {
    uint32_t s = 0u;
#pragma unroll
    for (int w = 0; w < NWAVES; ++w) s += whist[w * HIST_STRIDE + tid];
    hist[tid] = s;
  }
  __syncthreads();

  if (tid == 0) {
    uint32_t cum = 0u, b = 0u;
    for (; b < 256u; ++b) {
      const uint32_t c = hist[b];
      if (cum + c > (uint32_t)K_RANK) break;
      cum += c;
    }
    s_state[0] = b << 24;
    s_state[1] = (uint32_t)K_RANK - cum;
  }
  __syncthreads();

  uint32_t prefix = s_state[0];
  uint32_t pmask  = 0xFF000000u;
  int      k      = (int)s_state[1];

  // ---- Passes 1..3: shared histogram (mantissa bits spread evenly) --------
  for (int pass = 1; pass < 4; ++pass) {
    const int shift = 24 - 8 * pass;

    if (tid < 256) hist[tid] = 0u;
    __syncthreads();

#pragma unroll 4
    for (int j = 0; j < ITERS; ++j) {
      const int c = tid + j * BLOCK;
      const uint4 q = reinterpret_cast<uint4*>(raw)[c];
      uint32_t kx;
      kx = mono_key(q.x);
      if ((kx & pmask) == prefix) atomicAdd(&hist[(kx >> shift) & 255u], 1u);
      kx = mono_key(q.y);
      if ((kx & pmask) == prefix) atomicAdd(&hist[(kx >> shift) & 255u], 1u);
      kx = mono_key(q.z);
      if ((kx & pmask) == prefix) atomicAdd(&hist[(kx >> shift) & 255u], 1u);
      kx = mono_key(q.w);
      if ((kx & pmask) == prefix) atomicAdd(&hist[(kx >> shift) & 255u], 1u);
    }
    __syncthreads();

    if (tid == 0) {
      uint32_t cum = 0u, b = 0u;
      const uint32_t kk = (uint32_t)k;
      for (; b < 256u; ++b) {
        const uint32_t c = hist[b];
        if (cum + c > kk) break;
        cum += c;
      }
      s_state[0] = prefix | (b << shift);
      s_state[1] = kk - cum;
    }
    __syncthreads();

    prefix = s_state[0];
    k      = (int)s_state[1];
    pmask |= (0xFFu << shift);
  }

  // invert the monotonic map to recover the exact cutoff float
  const uint32_t cbits = (prefix & 0x80000000u) ? (prefix & 0x7FFFFFFFu)
                                                : ~prefix;
  const float cutoff = __uint_as_float(cbits);

  // ---- Stage 3: out = relu(x - cutoff); non-temporal streaming stores -----
#pragma unroll 4
  for (int j = 0; j < ITERS; ++j) {
    const int c = tid + j * BLOCK;
    const uint4 q = reinterpret_cast<uint4*>(raw)[c];
    v4f r;
    r.x = fmaxf(__uint_as_float(q.x) - cutoff, 0.0f);
    r.y = fmaxf(__uint_as_float(q.y) - cutoff, 0.0f);
    r.z = fmaxf(__uint_as_float(q.z) - cutoff, 0.0f);
    r.w = fmaxf(__uint_as_float(q.w) - cutoff, 0.0f);
    __builtin_nontemporal_store(r, reinterpret_cast<v4f*>(dst) + c);
  }
}

extern "C" void kernel_launch(void* const* d_in, const int* in_sizes, int n_in,
                              void* d_out, int out_size, void* d_ws, size_t ws_size,
                              hipStream_t stream) {
  (void)n_in; (void)out_size; (void)d_ws; (void)ws_size;
  const float* in = (const float*)d_in[0];
  float* out = (float*)d_out;

  const int rows = in_sizes[0] / N_COLS;  // 1024 for the reference shapes
  const size_t lds_bytes =
      (size_t)(N_COLS + NWAVES * HIST_STRIDE + 256 + 8) * sizeof(uint32_t);  // ~289KB

  xsrelu_perc_kernel<<<dim3(rows), dim3(BLOCK), lds_bytes, stream>>>(in, out);
}